// MIM_52192442581410
// MI455X (gfx1250) — compile-verified
//
#include <hip/hip_runtime.h>

typedef __attribute__((ext_vector_type(2))) float v2f;
typedef __attribute__((ext_vector_type(8))) float v8f;

#define B_   4
#define C_   128
#define H_   160
#define W_   160
#define HW_  (H_*W_)
#define CM_  64
#define CO_  32
#define OUTN 32
#define NS   64            // OUT*SR samples per axis

#define RT_S 132           // rT row stride (132 mod 64 == 4 -> conflict-free 64b A reads)
#define HT_S 68            // hT row stride (68 mod 64 == 4)
#define W1P_S 160          // w1x k-pair stride in dwords (160 mod 64 == 32 -> half-waves disjoint)
#define W2P_S 96           // w2x k-pair stride in dwords (96 mod 64 == 32)

#define WMMA_F32X4(A,Bv,Cc) \
  __builtin_amdgcn_wmma_f32_16x16x4_f32(false,(A),false,(Bv),(short)0,(Cc),false,false)

// ---------------- NCHW -> NHWC transpose (52 MB, one-shot, HBM bound ~5us) ---------
__global__ __launch_bounds__(256) void nchw2nhwc_k(const float* __restrict__ in,
                                                   float* __restrict__ out) {
  __shared__ float tb[32][33];
  const int b   = blockIdx.z;
  const int hw0 = blockIdx.x * 32;
  const int c0  = blockIdx.y * 32;
  const int tx  = threadIdx.x & 31;
  const int ty  = threadIdx.x >> 5;   // 0..7
#pragma unroll
  for (int r = 0; r < 4; ++r) {
    const int c = c0 + ty + r * 8;
    tb[ty + r * 8][tx] = in[(size_t)(b * C_ + c) * HW_ + hw0 + tx];
  }
  __syncthreads();
#pragma unroll
  for (int r = 0; r < 4; ++r) {
    const int hw = hw0 + ty + r * 8;
    out[((size_t)b * HW_ + hw) * C_ + c0 + tx] = tb[tx][ty + r * 8];
  }
}

// one bilinear tap: 128 contiguous fp32 channels, lane -> float4 (NHWC fast path)
template<bool NHWC>
__device__ __forceinline__ float4 tap_load(const float* __restrict__ fb, int yx, int lane) {
  if (NHWC) {
    const float4* p = (const float4*)(fb + (size_t)yx * C_);
    return p[lane];
  } else {
    const float* p = fb + (size_t)(lane * 4) * HW_ + yx;
    float4 v;
    v.x = p[0]; v.y = p[HW_]; v.z = p[2 * HW_]; v.w = p[3 * HW_];
    return v;
  }
}

// ---------------- fused ROI-align + GEMM1 + BN/ReLU + GEMM2, one block per ROI -----
template<bool NHWC>
__global__ __launch_bounds__(256) void roi_mlp_k(
    const float* __restrict__ feat, const float* __restrict__ boxes,
    const float* __restrict__ W1g, const float* __restrict__ b1g,
    const float* __restrict__ gamma, const float* __restrict__ beta,
    const float* __restrict__ mean, const float* __restrict__ var,
    const float* __restrict__ W2g, const float* __restrict__ b2g,
    float* __restrict__ out) {
  __shared__ float rT[128 * RT_S];          // roi-align tile  [128 pos][128 ch]
  __shared__ float hT[128 * HT_S];          // hidden tile     [128 pos][64 ch]
  __shared__ float w1x[(C_ / 2) * W1P_S];   // W1, k-pair interleaved for 64b B loads
  __shared__ float w2x[(CM_ / 2) * W2P_S];  // W2, k-pair interleaved
  __shared__ int   slo[2 * NS];
  __shared__ int   shi[2 * NS];
  __shared__ float sfr[2 * NS];
  __shared__ float sva[2 * NS];
  __shared__ float bnScale[CM_], bnShift[CM_], b2s[CO_];

  const int tid  = threadIdx.x;
  const int lane = tid & 31;
  const int wv   = tid >> 5;           // 8 waves
  const int k    = blockIdx.x;

  const int   bi    = (int)boxes[k * 5 + 0];
  const float x1    = boxes[k * 5 + 1], y1 = boxes[k * 5 + 2];
  const float x2    = boxes[k * 5 + 3], y2 = boxes[k * 5 + 4];
  const float roi_w = fmaxf(x2 - x1, 1.f);
  const float roi_h = fmaxf(y2 - y1, 1.f);

  // ---- bilinear index/weight tables (torchvision aligned=False, SR=2) ----
  if (tid < 2 * NS) {
    const bool  isx   = tid >= NS;
    const int   i     = isx ? tid - NS : tid;
    const float L     = isx ? (float)W_ : (float)H_;
    const float start = isx ? x1 : y1;
    const float scale = (isx ? roi_w : roi_h) * (1.f / OUTN);
    const float coord = start + ((float)i + 0.5f) * 0.5f * scale;
    const float valid = (coord >= -1.f && coord <= L) ? 1.f : 0.f;
    float c = fminf(fmaxf(coord, 0.f), L - 1.f);
    int lo = (int)floorf(c);
    if (lo > (int)L - 1) lo = (int)L - 1;
    int hi = lo + 1; if (hi > (int)L - 1) hi = (int)L - 1;
    slo[tid] = lo; shi[tid] = hi; sfr[tid] = c - (float)lo; sva[tid] = valid;
  }
  if (tid < CM_) {                                  // fold b1 + BN into scale/shift
    const float sc = gamma[tid] * rsqrtf(var[tid] + 1e-5f);
    bnScale[tid] = sc;
    bnShift[tid] = (b1g[tid] - mean[tid]) * sc + beta[tid];
  }
  if (tid >= 64 && tid < 64 + CO_) b2s[tid - 64] = b2g[tid - 64];
  // weights -> LDS, k-pair interleaved: fragment {W[k][c],W[k+1][c]} contiguous
  for (int idx = tid; idx < C_ * CM_; idx += 256) {
    const int kr = idx >> 6, c = idx & 63;
    w1x[(kr >> 1) * W1P_S + c * 2 + (kr & 1)] = W1g[idx];
  }
  for (int idx = tid; idx < CM_ * CO_; idx += 256) {
    const int kr = idx >> 5, c = idx & 31;
    w2x[(kr >> 1) * W2P_S + c * 2 + (kr & 1)] = W2g[idx];
  }
  __syncthreads();

  const float* fb = NHWC ? feat + (size_t)bi * HW_ * C_
                         : feat + (size_t)bi * C_ * HW_;
  const int koff  = (lane >> 4) << 1;              // WMMA A k-offset per half-wave
  const int kphi  = (lane >> 4);                   // k-pair offset per half-wave
  const int ncol  = lane & 15;
  const int bcol2 = ncol * 2;
  const int mb2   = wv * 16 + ((lane >> 4) << 3);  // D-fragment row base

  for (int t = 0; t < 8; ++t) {                    // 8 tiles of 4 output rows
    // ---- phase 1: ROI-align 16 positions per wave, lane = channel quad ----
    for (int i = 0; i < 16; ++i) {
      const int p  = wv * 16 + i;
      const int py = t * 4 + (p >> 5);
      const int px = p & 31;
      float4 acc; acc.x = acc.y = acc.z = acc.w = 0.f;
#pragma unroll
      for (int sy = 0; sy < 2; ++sy) {
        const int   yi = 2 * py + sy;
        const int   yl = slo[yi] * W_, yh = shi[yi] * W_;
        const float fy = sfr[yi], vy = sva[yi], wy0 = 1.f - sfr[yi];
#pragma unroll
        for (int sx = 0; sx < 2; ++sx) {
          const int   xi = NS + 2 * px + sx;
          const int   xl = slo[xi], xh = shi[xi];
          const float fx = sfr[xi];
          const float m  = vy * sva[xi] * 0.25f;   // validity * subsample-mean
          const float w00 = wy0 * (1.f - fx) * m, w01 = wy0 * fx * m;
          const float w10 = fy  * (1.f - fx) * m, w11 = fy  * fx * m;
          const float4 t00 = tap_load<NHWC>(fb, yl + xl, lane);
          const float4 t01 = tap_load<NHWC>(fb, yl + xh, lane);
          const float4 t10 = tap_load<NHWC>(fb, yh + xl, lane);
          const float4 t11 = tap_load<NHWC>(fb, yh + xh, lane);
          acc.x = fmaf(w00, t00.x, fmaf(w01, t01.x, fmaf(w10, t10.x, fmaf(w11, t11.x, acc.x))));
          acc.y = fmaf(w00, t00.y, fmaf(w01, t01.y, fmaf(w10, t10.y, fmaf(w11, t11.y, acc.y))));
          acc.z = fmaf(w00, t00.z, fmaf(w01, t01.z, fmaf(w10, t10.z, fmaf(w11, t11.z, acc.z))));
          acc.w = fmaf(w00, t00.w, fmaf(w01, t01.w, fmaf(w10, t10.w, fmaf(w11, t11.w, acc.w))));
        }
      }
      *(float4*)&rT[p * RT_S + (lane << 2)] = acc;
    }
    __syncthreads();

    // ---- phase 2: GEMM1 [128x128]x[128x64] via v_wmma_f32_16x16x4_f32 ----
    v8f accs[4] = {};
    {
      const int arow = (wv * 16 + ncol) * RT_S;
#pragma unroll 4
      for (int kk = 0; kk < 32; ++kk) {
        const v2f a = *(const v2f*)&rT[arow + kk * 4 + koff];
        const int brow = (kk * 2 + kphi) * W1P_S + bcol2;
#pragma unroll
        for (int n = 0; n < 4; ++n) {
          const v2f b = *(const v2f*)&w1x[brow + n * 32];
          accs[n] = WMMA_F32X4(a, b, accs[n]);
        }
      }
    }
    // BN + ReLU, stash h tile
#pragma unroll
    for (int n = 0; n < 4; ++n) {
      const int   d  = n * 16 + ncol;
      const float sc = bnScale[d], sh = bnShift[d];
#pragma unroll
      for (int v = 0; v < 8; ++v)
        hT[(mb2 + v) * HT_S + d] = fmaxf(fmaf(accs[n][v], sc, sh), 0.f);
    }
    __syncthreads();

    // ---- phase 3: GEMM2 [128x64]x[64x32] ----
    v8f od[2] = {};
    {
      const int arow = (wv * 16 + ncol) * HT_S;
#pragma unroll 4
      for (int kk = 0; kk < 16; ++kk) {
        const v2f a = *(const v2f*)&hT[arow + kk * 4 + koff];
        const int brow = (kk * 2 + kphi) * W2P_S + bcol2;
#pragma unroll
        for (int n = 0; n < 2; ++n) {
          const v2f b = *(const v2f*)&w2x[brow + n * 32];
          od[n] = WMMA_F32X4(a, b, od[n]);
        }
      }
    }
    // ---- store: output layout [K, Co, 32*32]; lane owns 8 consecutive pos ----
    {
      const int posg = t * 128 + mb2;
      float* ob = out + (size_t)k * (CO_ * OUTN * OUTN);
#pragma unroll
      for (int n = 0; n < 2; ++n) {
        const int   co   = n * 16 + ncol;
        const float bias = b2s[co];
        float* dst = ob + (size_t)co * (OUTN * OUTN) + posg;
        float4 lo, hi;
        lo.x = od[n][0] + bias; lo.y = od[n][1] + bias;
        lo.z = od[n][2] + bias; lo.w = od[n][3] + bias;
        hi.x = od[n][4] + bias; hi.y = od[n][5] + bias;
        hi.z = od[n][6] + bias; hi.w = od[n][7] + bias;
        *(float4*)(dst)     = lo;
        *(float4*)(dst + 4) = hi;
      }
    }
    __syncthreads();   // before next tile overwrites rT/hT
  }
}

extern "C" void kernel_launch(void* const* d_in, const int* in_sizes, int n_in,
                              void* d_out, int out_size, void* d_ws, size_t ws_size,
                              hipStream_t stream) {
  const float* x     = (const float*)d_in[0];
  const float* boxes = (const float*)d_in[1];
  const float* W1    = (const float*)d_in[2];
  const float* b1    = (const float*)d_in[3];
  const float* gam   = (const float*)d_in[4];
  const float* bet   = (const float*)d_in[5];
  const float* mean  = (const float*)d_in[6];
  const float* var   = (const float*)d_in[7];
  const float* W2    = (const float*)d_in[8];
  const float* b2    = (const float*)d_in[9];
  float* out = (float*)d_out;

  const int    K         = in_sizes[1] / 5;
  const size_t featBytes = (size_t)B_ * C_ * HW_ * sizeof(float);

  if (ws_size >= featBytes) {
    float* nhwc = (float*)d_ws;
    dim3 g(HW_ / 32, C_ / 32, B_);
    nchw2nhwc_k<<<g, 256, 0, stream>>>(x, nhwc);
    roi_mlp_k<true><<<K, 256, 0, stream>>>(nhwc, boxes, W1, b1, gam, bet, mean, var, W2, b2, out);
  } else {
    roi_mlp_k<false><<<K, 256, 0, stream>>>(x, boxes, W1, b1, gam, bet, mean, var, W2, b2, out);
  }
}